// FeatureInteraction_53721450938755
// MI455X (gfx1250) — compile-verified
//
#include <hip/hip_runtime.h>

typedef __attribute__((ext_vector_type(2))) float v2f;
typedef __attribute__((ext_vector_type(4))) float v4f;
typedef __attribute__((ext_vector_type(8))) float v8f;
typedef __attribute__((ext_vector_type(4))) unsigned int v4u;
typedef __attribute__((ext_vector_type(8))) int v8i;
typedef __attribute__((ext_vector_type(4))) int v4i;

#define CDIM  32
#define BDIM  16384
#define DDIM  64
#define NPAIR 528
#define LROW  68          // padded LDS row stride (floats): banks = 4*row + k, conflict-free
#define WAVES 4

__device__ __forceinline__ int triu_idx(int i, int j) {
    // offset of (i,j) in row-major upper triangle (incl. diagonal) of 32x32
    return i * 32 - (i * (i - 1)) / 2 + (j - i);
}

__global__ __launch_bounds__(WAVES * 32)
void gram_triu_wmma(const float* __restrict__ x, float* __restrict__ out) {
    __shared__ float lds_all[WAVES * CDIM * LROW];

    const int lane  = threadIdx.x & 31;
    const int wavei = __builtin_amdgcn_readfirstlane((int)(threadIdx.x >> 5));
    const int b     = blockIdx.x * WAVES + wavei;   // one batch per wave

    float* lds = lds_all + wavei * (CDIM * LROW);

    const int sub = lane >> 4;      // half-wave id: 0/1
    const int m   = lane & 15;      // 0..15

#if __has_builtin(__builtin_amdgcn_tensor_load_to_lds)
    // ---- TDM: one DMA per wave moves X_b[32][64] (rows 4 MiB apart) into LDS,
    //      with hardware padding producing the 68-DWORD conflict-free pitch.
    {
        const unsigned long long ga =
            (unsigned long long)(const void*)(x + (size_t)b * DDIM);
        const unsigned lds_byte = (unsigned)(wavei * (CDIM * LROW * 4));

        v4u g0;
        g0[0] = 1u;                                   // count=1, user descriptor
        g0[1] = lds_byte;                             // lds_addr
        g0[2] = (unsigned)(ga & 0xffffffffu);         // global_addr[31:0]
        g0[3] = (unsigned)((ga >> 32) & 0x1ffffffu)   // global_addr[56:32]
              | 0x80000000u;                          // type=2 ("image") in [127:126]

        v8i g1;
        g1[0] = (2 << 16)      // data_size = 4 bytes
              | (1 << 20)      // pad_enable
              | (5 << 22)      // pad_interval: 64 DWORDs (one 256B row)
              | (3 << 25);     // pad_amount:   4 DWORDs  -> 68-DWORD LDS pitch
        g1[1] = (int)(64u << 16);   // tensor_dim0 = 64 (low half)
        g1[2] = (int)(32u << 16);   // tensor_dim0 hi = 0 | tensor_dim1 = 32 (low half)
        g1[3] = (int)(64u << 16);   // tensor_dim1 hi = 0 | tile_dim0 = 64
        g1[4] = 32;                 // tile_dim1 = 32 | tile_dim2 = 0 (unused)
        g1[5] = 0x100000;           // tensor_dim0_stride = B*D = 1048576 elements
        g1[6] = 0;                  // stride0 hi | tensor_dim1_stride lo
        g1[7] = 0;                  // tensor_dim1_stride hi

        v4i g2;
        g2[0] = 1;                  // tensor_dim2 = 1 (benign)
        g2[1] = 1;                  // tensor_dim3 = 1 (benign)
        g2[2] = 0;                  // tensor_dim2_stride lo
        g2[3] = 0;                  // stride hi | tile_dim3 = 0 (unused)

        v4i g3;
        g3[0] = 0;                  // tensor_dim3_stride lo
        g3[1] = (int)(1u << 16);    // stride hi | tensor_dim4 = 1 (benign, low half)
        g3[2] = 0;                  // tensor_dim4 hi | tile_dim4 = 0 (unused)
        g3[3] = 0;

        v8i gz = {0, 0, 0, 0, 0, 0, 0, 0};   // extra group (clang-23 6-arg form)

        __builtin_amdgcn_tensor_load_to_lds(g0, g1, g2, g3, gz, 0);
        __builtin_amdgcn_s_wait_tensorcnt(0);
        // Escape the LDS object so the compiler cannot treat it as never-written:
        // the DMA (modeled by this asm's memory clobber) filled it.
        asm volatile("" : : "v"((void*)lds_all) : "memory");
    }
#else
    // ---- fallback: staged coalesced loads (each half-wave loads one 256B row)
    {
        const int col = m * 4;
        const float* xb = x + (size_t)b * DDIM;
#pragma unroll
        for (int it = 0; it < 16; ++it) {
            const int c = it * 2 + sub;
            const v4f v = __builtin_nontemporal_load(
                (const v4f*)(xb + (size_t)c * (size_t)(BDIM * DDIM) + col));
            *(v4f*)(lds + c * LROW + col) = v;
        }
        __syncthreads();
    }
#endif

    // ---- WMMA f32 16x16x4: a_i holds both the A-operand of X_i and the B-operand of X_i^T
    const float* r0 = lds + m * LROW + 2 * sub;           // rows 0..15  (tile 0)
    const float* r1 = lds + (m + 16) * LROW + 2 * sub;    // rows 16..31 (tile 1)

    v8f c00 = {}; v8f c01 = {}; v8f c11 = {};
#pragma unroll
    for (int kk = 0; kk < DDIM; kk += 4) {
        const v2f a0 = *(const v2f*)(r0 + kk);
        const v2f a1 = *(const v2f*)(r1 + kk);
        c00 = __builtin_amdgcn_wmma_f32_16x16x4_f32(false, a0, false, a0, (short)0, c00, false, false);
        c01 = __builtin_amdgcn_wmma_f32_16x16x4_f32(false, a0, false, a1, (short)0, c01, false, false);
        c11 = __builtin_amdgcn_wmma_f32_16x16x4_f32(false, a1, false, a1, (short)0, c11, false, false);
    }

    // ---- scatter upper triangle: lane owns (i = v+8*sub, j = m) of each tile
    float* outb = out + (size_t)b * NPAIR;
#pragma unroll
    for (int v = 0; v < 8; ++v) {
        const int i0 = v + 8 * sub;   // 0..15 row within tile
        const int j0 = m;             // 0..15 col within tile
        if (j0 >= i0)
            __builtin_nontemporal_store(c00[v], &outb[triu_idx(i0, j0)]);
        // tile01: i in [0,16), j in [16,32) -> always upper triangle
        __builtin_nontemporal_store(c01[v], &outb[triu_idx(i0, 16 + j0)]);
        if (j0 >= i0)
            __builtin_nontemporal_store(c11[v], &outb[triu_idx(16 + i0, 16 + j0)]);
    }
}

extern "C" void kernel_launch(void* const* d_in, const int* in_sizes, int n_in,
                              void* d_out, int out_size, void* d_ws, size_t ws_size,
                              hipStream_t stream) {
    const float* x = (const float*)d_in[0];
    float* out = (float*)d_out;
    dim3 grid(BDIM / WAVES);     // 4096 blocks, 4 waves each -> 16384 batches
    dim3 block(WAVES * 32);
    gram_triu_wmma<<<grid, block, 0, stream>>>(x, out);
}